// GraphbackboneWoGlobal_75265006895365
// MI455X (gfx1250) — compile-verified
//
#include <hip/hip_runtime.h>
#include <hip/hip_bf16.h>
#include <stdint.h>

// ---------------------------------------------------------------------------
// GraphbackboneWoGlobal: ragged->padded gather + off-by-one mask + DETR sine
// positional embedding.  Pure HBM-streaming problem (~419 MB total traffic,
// ~18us floor at 23.3 TB/s).  No matrix math -> no WMMA; use CDNA5 async
// global<->LDS DMA (ASYNCcnt) for the bulk feature copy + b128 vmem.
// 2D grid (x = row chunk, y = batch) avoids 64-bit div/mod in every block.
// ---------------------------------------------------------------------------

#define C_DIM  256   // channels per node
#define FEATS  128   // NUM_POS_FEATS
#define TWO_PI 6.28318530717958647692f
// log2(10000)/64  -> dim_t[f] = 2^(K * (f>>1)),  1/dim_t = exp2(-K*(f>>1))
#define KLOG2  0.20762050593046014f

#if __has_builtin(__builtin_amdgcn_global_load_async_to_lds_b128) && \
    __has_builtin(__builtin_amdgcn_global_store_async_from_lds_b128) && \
    __has_builtin(__builtin_amdgcn_s_wait_asynccnt)
#define USE_ASYNC_LDS 1
#else
#define USE_ASYNC_LDS 0
#endif

typedef int v4i __attribute__((__vector_size__(4 * sizeof(int))));
typedef __attribute__((address_space(3))) v4i* lds_v4i_t;  // LDS-typed pointer

// Low 32 bits of a flat LDS address are the LDS byte offset (LDS aperture
// keeps the base in addr[63:32]); inttoptr avoids addrspacecast diagnostics.
__device__ __forceinline__ lds_v4i_t to_lds(const void* p) {
    return (lds_v4i_t)(uintptr_t)(uint32_t)(uintptr_t)p;
}

// ---------------------------------------------------------------------------
// Feature gather (dominant stream: ~268 MB).  One wave32 per 1KB row
// (8 waves/block); blockIdx.y = batch, blockIdx.x = 8-row chunk.  Valid rows
// stream global -> LDS -> global through the async DMA engine (two b128 per
// lane each way); invalid rows store zeros directly.  n/b are wave-uniform
// so branches keep EXEC fully populated for the async ops; per-wave private
// LDS chunks need no barriers.
// ---------------------------------------------------------------------------
__global__ void feat_kernel(const float* __restrict__ pe,
                            const int* __restrict__ meta,
                            int N, int total,
                            float* __restrict__ out_feat) {
#if USE_ASYNC_LDS
    __shared__ __align__(16) float lds_buf[8 * C_DIM];   // 1KB per wave
#endif
    const int tid  = threadIdx.x;
    const int wave = tid >> 5;
    const int lane = tid & 31;
    const int b = blockIdx.y;
    const int n = blockIdx.x * 8 + wave;
    if (n >= N) return;

    const int offset = meta[3 * b + 0];
    const int n_eff  = meta[3 * b + 1];
    const bool valid = n < n_eff;
    const int  src   = min(offset + n, total - 1);
    const long long rid = (long long)b * N + n;

    const float4* src4 = (const float4*)pe + (long long)src * 64;
    float4*       dst4 = (float4*)out_feat + rid * 64;

    if (valid) {
#if USE_ASYNC_LDS
        float* lchunk = &lds_buf[wave * C_DIM];
        lds_v4i_t l0 = to_lds(lchunk + lane * 4);
        lds_v4i_t l1 = to_lds(lchunk + 128 + lane * 4);
        __builtin_amdgcn_global_load_async_to_lds_b128((v4i*)(src4 + lane),      l0, 0, 0);
        __builtin_amdgcn_global_load_async_to_lds_b128((v4i*)(src4 + 32 + lane), l1, 0, 0);
        __builtin_amdgcn_s_wait_asynccnt(0);
        __builtin_amdgcn_global_store_async_from_lds_b128((v4i*)(dst4 + lane),      l0, 0, 0);
        __builtin_amdgcn_global_store_async_from_lds_b128((v4i*)(dst4 + 32 + lane), l1, 0, 0);
        __builtin_amdgcn_s_wait_asynccnt(0);
#else
        __builtin_prefetch(src4 + lane, 0, 0);           // global_prefetch_b8
        dst4[lane]      = src4[lane];
        dst4[32 + lane] = src4[32 + lane];
#endif
    } else {
        const float4 z = {0.0f, 0.0f, 0.0f, 0.0f};
        dst4[lane]      = z;
        dst4[32 + lane] = z;
    }
}

// ---------------------------------------------------------------------------
// Per-batch metadata: offset[b] (exclusive scan of num_nodes), n_eff[b],
// num_nodes[b], and scale = 2pi/(max_coord + eps) per axis.
// ---------------------------------------------------------------------------
__global__ void prep_kernel(const int* __restrict__ num_nodes,
                            const float* __restrict__ centroid,
                            const int* __restrict__ max_node_p,
                            int total, int* __restrict__ meta,
                            float* __restrict__ scales) {
    __shared__ int   s_off[256];
    __shared__ float s_mx[256];
    __shared__ float s_my[256];
    const int b   = blockIdx.x;
    const int tid = threadIdx.x;
    const int maxn = max_node_p[0];

    int part = 0;
    for (int i = tid; i < b; i += 256) part += num_nodes[i];
    s_off[tid] = part;
    __syncthreads();
    for (int s = 128; s > 0; s >>= 1) {
        if (tid < s) s_off[tid] += s_off[tid + s];
        __syncthreads();
    }
    const int offset = s_off[0];
    const int nn     = num_nodes[b];
    const int n_eff  = min(nn, maxn);

    float mx = 0.0f, my = 0.0f;       // coords are uniform[0,1): 0 is a safe identity
    for (int j = tid; j < n_eff; j += 256) {
        const int src = min(offset + j, total - 1);
        mx = fmaxf(mx, centroid[2 * src + 0]);
        my = fmaxf(my, centroid[2 * src + 1]);
    }
    s_mx[tid] = mx; s_my[tid] = my;
    __syncthreads();
    for (int s = 128; s > 0; s >>= 1) {
        if (tid < s) {
            s_mx[tid] = fmaxf(s_mx[tid], s_mx[tid + s]);
            s_my[tid] = fmaxf(s_my[tid], s_my[tid + s]);
        }
        __syncthreads();
    }
    if (tid == 0) {
        meta[3 * b + 0] = offset;
        meta[3 * b + 1] = n_eff;
        meta[3 * b + 2] = nn;
        scales[2 * b + 0] = TWO_PI / (s_mx[0] + 1e-6f);   // x
        scales[2 * b + 1] = TWO_PI / (s_my[0] + 1e-6f);   // y
    }
}

// ---------------------------------------------------------------------------
// Mask + sine embedding.  blockIdx.y = batch, blockIdx.x = 4-row chunk;
// each row handled by 64 lanes, one float4 of output each.  Output per row:
// [0..127] = y emb, [128..255] = x emb;
// out[f even] = sin(coord*scale/dim_t[f]), odd = cos.  Invalid rows compute
// with coord = 0 -> the reference's [0,1,0,1,...] pattern (NOT zeros).
// Mask uses the reference's num_nodes+1 off-by-one.
// ---------------------------------------------------------------------------
__global__ void pos_mask_kernel(const float* __restrict__ centroid,
                                const int* __restrict__ meta,
                                const float* __restrict__ scales,
                                int N, int total,
                                float* __restrict__ out_mask,
                                float* __restrict__ out_pos) {
    const int tid  = threadIdx.x;
    const int row  = tid >> 6;        // 0..3
    const int lane = tid & 63;        // 0..63
    const int b = blockIdx.y;
    const int n = blockIdx.x * 4 + row;
    if (n >= N) return;
    const long long gid = (long long)b * N + n;   // row id in [0, B*N)

    const int offset = meta[3 * b + 0];
    const int n_eff  = meta[3 * b + 1];
    const int nn     = meta[3 * b + 2];
    const bool valid = n < n_eff;
    const int  src   = min(offset + n, total - 1);

    if (lane == 0)
        out_mask[gid] = (n >= nn + 1) ? 1.0f : 0.0f;

    // first 128 channels use y (centroid[.,1]); last 128 use x (centroid[.,0])
    float coord, scale;
    if (lane < 32) { coord = centroid[2 * src + 1]; scale = scales[2 * b + 1]; }
    else           { coord = centroid[2 * src + 0]; scale = scales[2 * b + 0]; }
    if (!valid) coord = 0.0f;
    const float s = coord * scale;

    const int fbase = (lane * 4) & 127;          // even
    const int p0    = fbase >> 1;                // pair index for k=0,1
    const float inv0 = exp2f(-KLOG2 * (float)p0);
    const float inv1 = exp2f(-KLOG2 * (float)(p0 + 1));
    const float a0 = s * inv0;
    const float a1 = s * inv1;

    float4 v;
    v.x = __sinf(a0);    // f = fbase   (even)
    v.y = __cosf(a0);    // f = fbase+1 (odd, same dim_t pair)
    v.z = __sinf(a1);    // f = fbase+2
    v.w = __cosf(a1);    // f = fbase+3
    ((float4*)out_pos)[gid * 64 + lane] = v;
}

// ---------------------------------------------------------------------------
extern "C" void kernel_launch(void* const* d_in, const int* in_sizes, int n_in,
                              void* d_out, int out_size, void* d_ws, size_t ws_size,
                              hipStream_t stream) {
    const float* pe        = (const float*)d_in[0];   // [total, 256]
    const float* centroid  = (const float*)d_in[1];   // [total, 2]
    const int*   num_nodes = (const int*)d_in[2];     // [B]
    const int*   max_node  = (const int*)d_in[3];     // scalar on device

    const int B     = in_sizes[2];
    const int total = in_sizes[0] / C_DIM;
    const long long BN = (long long)out_size / (2 * C_DIM + 1);  // B*N (out = BN*513)
    const int N     = (int)(BN / B);

    float* out      = (float*)d_out;
    float* out_feat = out;                    // [B,N,256]
    float* out_mask = out + BN * C_DIM;       // [B,N]
    float* out_pos  = out_mask + BN;          // [B,N,256]

    int*   meta   = (int*)d_ws;                                   // 3 ints / batch
    float* scales = (float*)((char*)d_ws + (size_t)3 * B * 4);    // 2 floats / batch

    prep_kernel<<<B, 256, 0, stream>>>(num_nodes, centroid, max_node, total, meta, scales);

    dim3 grid2((unsigned)((N + 3) / 4), (unsigned)B);
    pos_mask_kernel<<<grid2, 256, 0, stream>>>(centroid, meta, scales, N, total,
                                               out_mask, out_pos);

    dim3 grid3((unsigned)((N + 7) / 8), (unsigned)B);
    feat_kernel<<<grid3, 256, 0, stream>>>(pe, meta, N, total, out_feat);
}